// WLNet_23527830847655
// MI455X (gfx1250) — compile-verified
//
#include <hip/hip_runtime.h>
#include <hip/hip_bf16.h>

// WLNet message passing on gfx1250 via V_WMMA_F32_16X16X4_F32 (full fp32).
// v3: data reuse. Gather kernels process 4 atoms/block (one B fragment feeds
// 4 WMMAs); row GEMMs use 2x2 wave tiling (4 accumulators). Branch-free inner
// loops: clamped addresses + register selects only.

#define Bsz   256
#define Asz   100
#define NBq   10
#define BONDq 110
#define AFq   82
#define BFq   6
#define Hq    300
#define KPAD  308            // 300 (h part) + 6 (bond part) + 2 pad
#define NT    19             // ceil(300/16) N tiles
#define NP    10             // N tile-pairs (2x2 tiling)
#define MTOT  (Bsz*Asz)      // 25600 rows
#define MT    (MTOT/16)      // 1600 M tiles
#define MP    (MT/2)         // 800 M tile-pairs
#define ATPB  4              // atoms per block in gather kernels
#define PANEL (16*KPAD)      // floats per atom A-panel

typedef float v2f __attribute__((ext_vector_type(2)));
typedef float v8f __attribute__((ext_vector_type(8)));

static __device__ __forceinline__ v8f wmma4(v2f a, v2f b, v8f c) {
  // D = A(16x4 f32) * B(4x16 f32) + C(16x16 f32)
  return __builtin_amdgcn_wmma_f32_16x16x4_f32(
      false, a, false, b, (short)0, c, false, false);
}

// ---------------------------------------------------------------------------
// K1: h = relu(atom_feats @ w_fc1)   [25600,82] x [82,300], 2x2 wave tiles
// ---------------------------------------------------------------------------
__global__ __launch_bounds__(256) void fc1_kernel(
    const float* __restrict__ af, const float* __restrict__ w,
    float* __restrict__ hout) {
  int tid = blockIdx.x * 8 + (threadIdx.x >> 5);
  if (tid >= MP * NP) return;
  const int lane = threadIdx.x & 31;
  const int m = lane & 15, half = lane >> 4, koff = half << 1;
  const int mp = tid / NP, np = tid - mp * NP;
  const int n0 = np * 32 + m, n1 = n0 + 16;
  const int row0 = mp * 32 + m, row1 = row0 + 16;
  const int nc0 = (n0 < Hq) ? n0 : (Hq - 1);
  const int nc1 = (n1 < Hq) ? n1 : (Hq - 1);

  const float* ap0 = af + row0 * AFq + koff;
  const float* ap1 = af + row1 * AFq + koff;
  const float* wp0 = w + koff * Hq + nc0;
  const float* wp1 = w + koff * Hq + nc1;
  v8f c00 = {}, c01 = {}, c10 = {}, c11 = {};
#pragma unroll 2
  for (int kk = 0; kk < 80; kk += 4) {          // k0 max 79 < 82, unguarded
    v2f a0 = *(const v2f*)ap0;
    v2f a1 = *(const v2f*)ap1;
    v2f b0; b0.x = wp0[0]; b0.y = wp0[Hq];
    v2f b1; b1.x = wp1[0]; b1.y = wp1[Hq];
    ap0 += 4; ap1 += 4; wp0 += 4 * Hq; wp1 += 4 * Hq;
    c00 = wmma4(a0, b0, c00); c01 = wmma4(a0, b1, c01);
    c10 = wmma4(a1, b0, c10); c11 = wmma4(a1, b1, c11);
  }
  {                                             // tail: k0 in {80,81,82,83}
    const int k0 = 80 + koff;
    const int k0c = (k0 < AFq) ? k0 : AFq - 1;
    const int k1c = (k0 + 1 < AFq) ? k0 + 1 : AFq - 1;
    v2f a0, a1, b0, b1;
    float a0x = af[row0 * AFq + k0c], a0y = af[row0 * AFq + k1c];
    float a1x = af[row1 * AFq + k0c], a1y = af[row1 * AFq + k1c];
    a0.x = (k0 < AFq) ? a0x : 0.f;  a0.y = (k0 + 1 < AFq) ? a0y : 0.f;
    a1.x = (k0 < AFq) ? a1x : 0.f;  a1.y = (k0 + 1 < AFq) ? a1y : 0.f;
    b0.x = w[k0c * Hq + nc0]; b0.y = w[k1c * Hq + nc0];  // A cols zero -> any B
    b1.x = w[k0c * Hq + nc1]; b1.y = w[k1c * Hq + nc1];
    c00 = wmma4(a0, b0, c00); c01 = wmma4(a0, b1, c01);
    c10 = wmma4(a1, b0, c10); c11 = wmma4(a1, b1, c11);
  }
  const int rb0 = mp * 32 + half * 8, rb1 = rb0 + 16;
#pragma unroll
  for (int v = 0; v < 8; ++v) {
    if (n0 < Hq) {
      hout[(rb0 + v) * Hq + n0] = fmaxf(c00[v], 0.f);
      hout[(rb1 + v) * Hq + n0] = fmaxf(c10[v], 0.f);
    }
    if (n1 < Hq) {
      hout[(rb0 + v) * Hq + n1] = fmaxf(c01[v], 0.f);
      hout[(rb1 + v) * Hq + n1] = fmaxf(c11[v], 0.f);
    }
  }
}

// ---------------------------------------------------------------------------
// K2: nei[ba,:] = sum_j mask_j * relu(h[ag[ba,j]] @ Wn[:300] +
//                                     bond[bg[ba,j]] @ Wn[300:306] + b_nei)
// 4 atoms/block: one B fragment per k-step feeds 4 WMMAs. A-panels (10
// gathered h rows + bond feats, zero-padded to 16x308) staged in LDS.
// Neighbor reduction straight from the C fragment layout + shfl_xor(16).
// ---------------------------------------------------------------------------
__global__ __launch_bounds__(256) void nei_kernel(
    const float* __restrict__ h, const float* __restrict__ bond,
    const int* __restrict__ ag, const int* __restrict__ bg,
    const int* __restrict__ nnb, const float* __restrict__ wn,
    const float* __restrict__ bn, float* __restrict__ out) {
  __shared__ float Ap[ATPB * PANEL];            // 78,848 B
  __shared__ int sib[ATPB * NBq], sia[ATPB * NBq];
  __shared__ int sbb[ATPB * NBq], sbe[ATPB * NBq];
  const int ba0 = blockIdx.x * ATPB;
  if (threadIdx.x < ATPB * NBq) {
    int q = threadIdx.x / NBq, r = threadIdx.x - q * NBq;
    int e = ((ba0 + q) * NBq + r) * 2;
    sib[threadIdx.x] = ag[e + 0]; sia[threadIdx.x] = ag[e + 1];
    sbb[threadIdx.x] = bg[e + 0]; sbe[threadIdx.x] = bg[e + 1];
  }
  __syncthreads();
  for (int idx = threadIdx.x; idx < ATPB * PANEL; idx += 256) {
    int q = idx / PANEL, rem = idx - q * PANEL;
    int r = rem / KPAD, c = rem - r * KPAD;
    float v = 0.f;
    if (r < NBq) {
      int s = q * NBq + r;
      if (c < Hq)            v = h[(sib[s] * Asz + sia[s]) * Hq + c];
      else if (c < Hq + BFq) v = bond[(sbb[s] * BONDq + sbe[s]) * BFq + (c - Hq)];
    }
    Ap[idx] = v;
  }
  __syncthreads();
  const int wave = threadIdx.x >> 5, lane = threadIdx.x & 31;
  const int m = lane & 15, half = lane >> 4, koff = half << 1;
  const float* apl = Ap + m * KPAD + koff;
  for (int t = wave; t < NT; t += 8) {
    const int n = t * 16 + m;
    const int nc = (n < Hq) ? n : (Hq - 1);
    const float* wp = wn + koff * Hq + nc;
    v8f acc[ATPB] = {};
#pragma unroll 2
    for (int kk = 0; kk < 304; kk += 4) {       // k0 max 303 < 306, unguarded
      v2f b; b.x = wp[0]; b.y = wp[Hq];
      wp += 4 * Hq;
#pragma unroll
      for (int q = 0; q < ATPB; ++q) {
        v2f a = *(const v2f*)(apl + q * PANEL + kk);
        acc[q] = wmma4(a, b, acc[q]);
      }
    }
    {                                           // tail: k0 in {304..307}
      const int k0 = 304 + koff;
      v2f b;                                    // LDS cols 306/307 are zero
      b.x = wn[(k0     < Hq + BFq ? k0     : Hq + BFq - 1) * Hq + nc];
      b.y = wn[(k0 + 1 < Hq + BFq ? k0 + 1 : Hq + BFq - 1) * Hq + nc];
#pragma unroll
      for (int q = 0; q < ATPB; ++q) {
        v2f a = *(const v2f*)(apl + q * PANEL + 304);
        acc[q] = wmma4(a, b, acc[q]);
      }
    }
    const float bias = bn[nc];
#pragma unroll
    for (int q = 0; q < ATPB; ++q) {
      const int nn = nnb[ba0 + q];
      float s = 0.f;
#pragma unroll
      for (int v = 0; v < 8; ++v) {
        int rj = v + half * 8;
        float x = fmaxf(acc[q][v] + bias, 0.f);
        s += (rj < nn) ? x : 0.f;               // nn <= 10; rows 10..15 padded
      }
      s += __shfl_xor(s, 16, 32);
      if (half == 0 && n < Hq) out[(ba0 + q) * Hq + n] = s;
    }
  }
}

// ---------------------------------------------------------------------------
// K3: h_new = relu(h @ Wa[:300] + nei @ Wa[300:600] + b_atom), 2x2 tiles
// ---------------------------------------------------------------------------
__global__ __launch_bounds__(256) void atom_kernel(
    const float* __restrict__ h, const float* __restrict__ nei,
    const float* __restrict__ wa, const float* __restrict__ bb,
    float* __restrict__ hout) {
  int tid = blockIdx.x * 8 + (threadIdx.x >> 5);
  if (tid >= MP * NP) return;
  const int lane = threadIdx.x & 31;
  const int m = lane & 15, half = lane >> 4, koff = half << 1;
  const int mp = tid / NP, np = tid - mp * NP;
  const int n0 = np * 32 + m, n1 = n0 + 16;
  const int row0 = mp * 32 + m, row1 = row0 + 16;
  const int nc0 = (n0 < Hq) ? n0 : (Hq - 1);
  const int nc1 = (n1 < Hq) ? n1 : (Hq - 1);
  v8f c00 = {}, c01 = {}, c10 = {}, c11 = {};
  {
    const float* ap0 = h + row0 * Hq + koff;
    const float* ap1 = h + row1 * Hq + koff;
    const float* wp0 = wa + koff * Hq + nc0;
    const float* wp1 = wa + koff * Hq + nc1;
#pragma unroll 3
    for (int kk = 0; kk < Hq; kk += 4) {        // K = 300 exact
      v2f a0 = *(const v2f*)ap0;
      v2f a1 = *(const v2f*)ap1;
      v2f b0; b0.x = wp0[0]; b0.y = wp0[Hq];
      v2f b1; b1.x = wp1[0]; b1.y = wp1[Hq];
      ap0 += 4; ap1 += 4; wp0 += 4 * Hq; wp1 += 4 * Hq;
      c00 = wmma4(a0, b0, c00); c01 = wmma4(a0, b1, c01);
      c10 = wmma4(a1, b0, c10); c11 = wmma4(a1, b1, c11);
    }
  }
  {
    const float* ap0 = nei + row0 * Hq + koff;
    const float* ap1 = nei + row1 * Hq + koff;
    const float* wp0 = wa + (Hq + koff) * Hq + nc0;
    const float* wp1 = wa + (Hq + koff) * Hq + nc1;
#pragma unroll 3
    for (int kk = 0; kk < Hq; kk += 4) {
      v2f a0 = *(const v2f*)ap0;
      v2f a1 = *(const v2f*)ap1;
      v2f b0; b0.x = wp0[0]; b0.y = wp0[Hq];
      v2f b1; b1.x = wp1[0]; b1.y = wp1[Hq];
      ap0 += 4; ap1 += 4; wp0 += 4 * Hq; wp1 += 4 * Hq;
      c00 = wmma4(a0, b0, c00); c01 = wmma4(a0, b1, c01);
      c10 = wmma4(a1, b0, c10); c11 = wmma4(a1, b1, c11);
    }
  }
  const float bias0 = bb[nc0], bias1 = bb[nc1];
  const int rb0 = mp * 32 + half * 8, rb1 = rb0 + 16;
#pragma unroll
  for (int v = 0; v < 8; ++v) {
    if (n0 < Hq) {
      hout[(rb0 + v) * Hq + n0] = fmaxf(c00[v] + bias0, 0.f);
      hout[(rb1 + v) * Hq + n0] = fmaxf(c10[v] + bias0, 0.f);
    }
    if (n1 < Hq) {
      hout[(rb0 + v) * Hq + n1] = fmaxf(c01[v] + bias1, 0.f);
      hout[(rb1 + v) * Hq + n1] = fmaxf(c11[v] + bias1, 0.f);
    }
  }
}

// ---------------------------------------------------------------------------
// K4: neiprod[ba,:] = sum_j mask_j * (h[ag] @ w2a) * (bond[bg] @ w2b)
// 4 atoms/block, same staging as K2.
// ---------------------------------------------------------------------------
__global__ __launch_bounds__(256) void nprod_kernel(
    const float* __restrict__ h, const float* __restrict__ bond,
    const int* __restrict__ ag, const int* __restrict__ bg,
    const int* __restrict__ nnb, const float* __restrict__ w2a,
    const float* __restrict__ w2b, float* __restrict__ out) {
  __shared__ float Ap[ATPB * PANEL];
  __shared__ int sib[ATPB * NBq], sia[ATPB * NBq];
  __shared__ int sbb[ATPB * NBq], sbe[ATPB * NBq];
  const int ba0 = blockIdx.x * ATPB;
  if (threadIdx.x < ATPB * NBq) {
    int q = threadIdx.x / NBq, r = threadIdx.x - q * NBq;
    int e = ((ba0 + q) * NBq + r) * 2;
    sib[threadIdx.x] = ag[e + 0]; sia[threadIdx.x] = ag[e + 1];
    sbb[threadIdx.x] = bg[e + 0]; sbe[threadIdx.x] = bg[e + 1];
  }
  __syncthreads();
  for (int idx = threadIdx.x; idx < ATPB * PANEL; idx += 256) {
    int q = idx / PANEL, rem = idx - q * PANEL;
    int r = rem / KPAD, c = rem - r * KPAD;
    float v = 0.f;
    if (r < NBq) {
      int s = q * NBq + r;
      if (c < Hq)            v = h[(sib[s] * Asz + sia[s]) * Hq + c];
      else if (c < Hq + BFq) v = bond[(sbb[s] * BONDq + sbe[s]) * BFq + (c - Hq)];
    }
    Ap[idx] = v;
  }
  __syncthreads();
  const int wave = threadIdx.x >> 5, lane = threadIdx.x & 31;
  const int m = lane & 15, half = lane >> 4, koff = half << 1;
  const float* apl = Ap + m * KPAD + koff;
  for (int t = wave; t < NT; t += 8) {
    const int n = t * 16 + m;
    const int nc = (n < Hq) ? n : (Hq - 1);
    v8f accA[ATPB] = {}, accB[ATPB] = {};
    {
      const float* wp = w2a + koff * Hq + nc;
#pragma unroll 3
      for (int kk = 0; kk < Hq; kk += 4) {      // K = 300 exact
        v2f b; b.x = wp[0]; b.y = wp[Hq];
        wp += 4 * Hq;
#pragma unroll
        for (int q = 0; q < ATPB; ++q) {
          v2f a = *(const v2f*)(apl + q * PANEL + kk);
          accA[q] = wmma4(a, b, accA[q]);
        }
      }
    }
#pragma unroll
    for (int kk = 0; kk < 8; kk += 4) {         // bond: K=6 padded to 8
      const int k0 = kk + koff;
      v2f b;                                    // LDS cols 306/307 are zero
      b.x = w2b[(k0     < BFq ? k0     : BFq - 1) * Hq + nc];
      b.y = w2b[(k0 + 1 < BFq ? k0 + 1 : BFq - 1) * Hq + nc];
#pragma unroll
      for (int q = 0; q < ATPB; ++q) {
        v2f a = *(const v2f*)(apl + q * PANEL + Hq + kk);
        accB[q] = wmma4(a, b, accB[q]);
      }
    }
#pragma unroll
    for (int q = 0; q < ATPB; ++q) {
      const int nn = nnb[ba0 + q];
      float s = 0.f;
#pragma unroll
      for (int v = 0; v < 8; ++v) {
        int rj = v + half * 8;
        s += (rj < nn) ? accA[q][v] * accB[q][v] : 0.f;
      }
      s += __shfl_xor(s, 16, 32);
      if (half == 0 && n < Hq) out[(ba0 + q) * Hq + n] = s;
    }
  }
}

// ---------------------------------------------------------------------------
// K5: local = (h @ w_fc2) * neiprod * (a < n_atoms[b]), 2x2 tiles, K=300
// ---------------------------------------------------------------------------
__global__ __launch_bounds__(256) void out_kernel(
    const float* __restrict__ h, const float* __restrict__ w,
    const float* __restrict__ nprod, const int* __restrict__ na,
    float* __restrict__ out) {
  int tid = blockIdx.x * 8 + (threadIdx.x >> 5);
  if (tid >= MP * NP) return;
  const int lane = threadIdx.x & 31;
  const int m = lane & 15, half = lane >> 4, koff = half << 1;
  const int mp = tid / NP, np = tid - mp * NP;
  const int n0 = np * 32 + m, n1 = n0 + 16;
  const int row0 = mp * 32 + m, row1 = row0 + 16;
  const int nc0 = (n0 < Hq) ? n0 : (Hq - 1);
  const int nc1 = (n1 < Hq) ? n1 : (Hq - 1);
  v8f c00 = {}, c01 = {}, c10 = {}, c11 = {};
  const float* ap0 = h + row0 * Hq + koff;
  const float* ap1 = h + row1 * Hq + koff;
  const float* wp0 = w + koff * Hq + nc0;
  const float* wp1 = w + koff * Hq + nc1;
#pragma unroll 3
  for (int kk = 0; kk < Hq; kk += 4) {
    v2f a0 = *(const v2f*)ap0;
    v2f a1 = *(const v2f*)ap1;
    v2f b0; b0.x = wp0[0]; b0.y = wp0[Hq];
    v2f b1; b1.x = wp1[0]; b1.y = wp1[Hq];
    ap0 += 4; ap1 += 4; wp0 += 4 * Hq; wp1 += 4 * Hq;
    c00 = wmma4(a0, b0, c00); c01 = wmma4(a0, b1, c01);
    c10 = wmma4(a1, b0, c10); c11 = wmma4(a1, b1, c11);
  }
  const int rb0 = mp * 32 + half * 8, rb1 = rb0 + 16;
#pragma unroll
  for (int v = 0; v < 8; ++v) {
    int r0 = rb0 + v, r1 = rb1 + v;
    int b0i = r0 / Asz, a0i = r0 - b0i * Asz;
    int b1i = r1 / Asz, a1i = r1 - b1i * Asz;
    bool m0 = (a0i < na[b0i]), m1 = (a1i < na[b1i]);
    if (n0 < Hq) {
      out[r0 * Hq + n0] = m0 ? c00[v] * nprod[r0 * Hq + n0] : 0.f;
      out[r1 * Hq + n0] = m1 ? c10[v] * nprod[r1 * Hq + n0] : 0.f;
    }
    if (n1 < Hq) {
      out[r0 * Hq + n1] = m0 ? c01[v] * nprod[r0 * Hq + n1] : 0.f;
      out[r1 * Hq + n1] = m1 ? c11[v] * nprod[r1 * Hq + n1] : 0.f;
    }
  }
}

// ---------------------------------------------------------------------------
extern "C" void kernel_launch(void* const* d_in, const int* in_sizes, int n_in,
                              void* d_out, int out_size, void* d_ws, size_t ws_size,
                              hipStream_t stream) {
  const float* atom_feats = (const float*)d_in[0];
  const float* bond_feats = (const float*)d_in[1];
  const float* w_fc1      = (const float*)d_in[2];
  const float* w_nei      = (const float*)d_in[3];
  const float* b_nei      = (const float*)d_in[4];
  const float* w_atom     = (const float*)d_in[5];
  const float* b_atom     = (const float*)d_in[6];
  const float* w2a        = (const float*)d_in[7];
  const float* w2b        = (const float*)d_in[8];
  const float* w_fc2      = (const float*)d_in[9];
  const int*   atom_graph = (const int*)d_in[10];
  const int*   bond_graph = (const int*)d_in[11];
  const int*   num_nbs    = (const int*)d_in[12];
  const int*   n_atoms    = (const int*)d_in[13];
  float* out = (float*)d_out;

  const size_t nH = (size_t)Bsz * Asz * Hq;   // 7,680,000 floats
  float* h_a = (float*)d_ws;                  // ws needs ~92.2 MB
  float* h_b = h_a + nH;
  float* nei = h_b + nH;                      // reused as neiprod at the end

  dim3 blk(256);
  const int gemm_blocks = (MP * NP + 7) / 8;      // 1000 blocks, 8 waves each
  const int gather_blocks = (Bsz * Asz) / ATPB;   // 6400 blocks

  fc1_kernel<<<gemm_blocks, blk, 0, stream>>>(atom_feats, w_fc1, h_a);

  // layer 0
  nei_kernel<<<gather_blocks, blk, 0, stream>>>(h_a, bond_feats, atom_graph,
                                                bond_graph, num_nbs, w_nei, b_nei, nei);
  atom_kernel<<<gemm_blocks, blk, 0, stream>>>(h_a, nei, w_atom, b_atom, h_b);
  // layer 1
  nei_kernel<<<gather_blocks, blk, 0, stream>>>(h_b, bond_feats, atom_graph,
                                                bond_graph, num_nbs, w_nei, b_nei, nei);
  atom_kernel<<<gemm_blocks, blk, 0, stream>>>(h_b, nei, w_atom, b_atom, h_a);
  // final layer
  nprod_kernel<<<gather_blocks, blk, 0, stream>>>(h_a, bond_feats, atom_graph,
                                                  bond_graph, num_nbs, w2a, w2b, nei);
  out_kernel<<<gemm_blocks, blk, 0, stream>>>(h_a, w_fc2, nei, n_atoms, out);
}